// RelGraphConvHeteroEmbed_61478161874990
// MI455X (gfx1250) — compile-verified
//
#include <hip/hip_runtime.h>

// ---------------------------------------------------------------------------
// RelGraphConvHeteroEmbed for MI455X (gfx1250, wave32)
//   h_A = scatter_add(embed1[src1], dst1) + scatter_add(embed2[src2], dst2)
//   h_B = scatter_add(embed0[src0], dst0) + scatter_add(embed3[src3], dst3)
//   out = relu(h + bias), out shape [2, N, 64] f32
// Memory-bound: gathers stream 512MB (embed tables are L2-resident after first
// touch), scatter resolves as global_atomic_add_f32 in L2 (51MB output << 192MB
// L2). Epilogue bias-broadcast-add done with V_WMMA_F32_16X16X4_F32.
// ---------------------------------------------------------------------------

typedef __attribute__((ext_vector_type(2))) float v2f;
typedef __attribute__((ext_vector_type(8))) float v8f;

#define EMBED_D 64

// ---- kernel 1: zero the accumulator (d_out is poisoned by the harness) ----
__global__ void zero_out_kernel(float4* __restrict__ out, int n4) {
    int i = blockIdx.x * blockDim.x + threadIdx.x;
    if (i < n4) out[i] = make_float4(0.f, 0.f, 0.f, 0.f);
}

// ---- kernel 2: wave-per-edge gather + L2 atomic scatter-add ----
// Each wave (32 lanes) handles one edge: lanes load float2 (256B coalesced row)
// and issue 2x global_atomic_add_f32 into the destination row.
// Prefetch the next edge's source row to hide random-gather latency.
__global__ void scatter_add_kernel(const float* __restrict__ embed,
                                   const int*   __restrict__ src,
                                   const int*   __restrict__ dst,
                                   float*       __restrict__ out,
                                   int nedges) {
    const int lane   = threadIdx.x & 31;
    const int wave   = (blockIdx.x * blockDim.x + threadIdx.x) >> 5;
    const int nwaves = (gridDim.x * blockDim.x) >> 5;

    for (int e = wave; e < nedges; e += nwaves) {
        const int s = src[e];
        const int d = dst[e];

        // software prefetch of next iteration's source row -> global_prefetch_b8
        const int e2 = e + nwaves;
        if (e2 < nedges) {
            const int s2 = src[e2];
            __builtin_prefetch(embed + (size_t)s2 * EMBED_D + lane * 2, 0, 0);
        }

        const v2f* sp = (const v2f*)(embed + (size_t)s * EMBED_D);
        const v2f  v  = sp[lane];                       // global_load_b64

        float* op = out + (size_t)d * EMBED_D + lane * 2;
        unsafeAtomicAdd(op,     v.x);                   // global_atomic_add_f32
        unsafeAtomicAdd(op + 1, v.y);
    }
}

// ---- kernel 3: bias-broadcast add + ReLU via V_WMMA_F32_16X16X4_F32 ----
// Tile = 16 node-rows x 16 feature-cols of the [2*N, 64] output.
// A (16x4): column K=0 all ones, rest zero  -> A x B broadcasts B's row 0.
// B (4x16): row K=0 = bias chunk, rest zero.
// D = A x B + C == C + bias  (every product term exact: 1*bias or 0*x),
// bit-identical to scalar adds. C/D VGPR layout per ISA 7.12.2.
__global__ void bias_relu_wmma_kernel(float* __restrict__ out,
                                      const float* __restrict__ bias,
                                      int total_tiles) {
    const int lane = threadIdx.x & 31;
    const int tile = blockIdx.x * (blockDim.x >> 5) + (threadIdx.x >> 5);
    if (tile >= total_tiles) return;   // wave-uniform: EXEC stays all-ones

    const int  ct = tile & 3;          // 4 column tiles of 16 (D=64)
    const long rt = (long)(tile >> 2); // row tile over 2*N rows
    const int  c0 = ct * 16;

    // A: ones in K=0 (VGPR0 lanes 0-15 hold M=0..15,K=0 per ISA layout)
    v2f a; a.x = (lane < 16) ? 1.0f : 0.0f; a.y = 0.0f;
    // B: bias chunk in K=0 row (VGPR0 lanes 0-15 hold K=0, N=0..15)
    v2f b; b.x = (lane < 16) ? bias[c0 + (lane & 15)] : 0.0f; b.y = 0.0f;

    // C/D layout: VGPR v, lane l -> row (v + 8*(l>=16)), col (l%16)
    const int  col   = c0 + (lane & 15);
    const long rbase = rt * 16 + ((lane >> 4) << 3);
    float* p = out + rbase * EMBED_D + col;

    v8f c;
#pragma unroll
    for (int v = 0; v < 8; ++v) c[v] = p[(long)v * EMBED_D];

    v8f d = __builtin_amdgcn_wmma_f32_16x16x4_f32(
        /*neg_a=*/false, a, /*neg_b=*/false, b,
        /*c_mod=*/(short)0, c, /*reuse_a=*/false, /*reuse_b=*/false);

#pragma unroll
    for (int v = 0; v < 8; ++v) {
        const float x = d[v];
        p[(long)v * EMBED_D] = x > 0.0f ? x : 0.0f;   // relu
    }
}

extern "C" void kernel_launch(void* const* d_in, const int* in_sizes, int n_in,
                              void* d_out, int out_size, void* d_ws, size_t ws_size,
                              hipStream_t stream) {
    const float* embed0 = (const float*)d_in[0];
    const float* embed1 = (const float*)d_in[1];
    const float* embed2 = (const float*)d_in[2];
    const float* embed3 = (const float*)d_in[3];
    const float* h_bias = (const float*)d_in[4];
    const int* src0 = (const int*)d_in[5];
    const int* dst0 = (const int*)d_in[6];
    const int* src1 = (const int*)d_in[7];
    const int* dst1 = (const int*)d_in[8];
    const int* src2 = (const int*)d_in[9];
    const int* dst2 = (const int*)d_in[10];
    const int* src3 = (const int*)d_in[11];
    const int* dst3 = (const int*)d_in[12];

    float* out = (float*)d_out;
    const int N = in_sizes[0] / EMBED_D;   // 100000
    const int E = in_sizes[5];             // 500000

    float* hA = out;                        // out[0] = h_A
    float* hB = out + (size_t)N * EMBED_D;  // out[1] = h_B

    // 1) zero accumulators
    const int n4 = out_size / 4;
    zero_out_kernel<<<(n4 + 255) / 256, 256, 0, stream>>>((float4*)out, n4);

    // 2) four scatter-add passes (atomics make concurrent dst updates safe;
    //    stream order guarantees all complete before the epilogue)
    const int blocks = 2048;  // 16384 waves; ~30 edges per wave
    scatter_add_kernel<<<blocks, 256, 0, stream>>>(embed1, src1, dst1, hA, E);
    scatter_add_kernel<<<blocks, 256, 0, stream>>>(embed2, src2, dst2, hA, E);
    scatter_add_kernel<<<blocks, 256, 0, stream>>>(embed0, src0, dst0, hB, E);
    scatter_add_kernel<<<blocks, 256, 0, stream>>>(embed3, src3, dst3, hB, E);

    // 3) bias + relu epilogue with f32 WMMA (2*N rows, 4 column tiles)
    const int total_tiles = (2 * N / 16) * 4;   // 50000, exact full waves
    const int waves_per_block = 256 / 32;
    bias_relu_wmma_kernel<<<(total_tiles + waves_per_block - 1) / waves_per_block,
                            256, 0, stream>>>(out, h_bias, total_tiles);
}